// RBFlayer_89026082111666
// MI455X (gfx1250) — compile-verified
//
#include <hip/hip_runtime.h>

// RBF layer: out[b,s,k] = exp(-||x_bs - c_k||^2 / sigma_k^2)
// B=32 S=1024 F=3 K=300. Output-store bound (39.3 MB @ 23.3 TB/s ~ 1.7us floor).
// Distance computed in ONE v_wmma_f32_16x16x4_f32 per 16x16 tile by augmenting:
//   A row = [x0,x1,x2,1], B col = [-2c0,-2c1,-2c2,||c||^2], C = ||x||^2 broadcast.
// Round 2: fast v_rcp_f32 instead of IEEE divide chain; loop split into 18 full
// tiles (full EXEC, fully unrolled, immediate-offset stores) + 1 predicated tail.

typedef __attribute__((ext_vector_type(2))) float v2f;
typedef __attribute__((ext_vector_type(8))) float v8f;

#define F_DIM 3
#define K_DIM 300
#define ROWS  (32 * 1024)       // B*S = 32768 rows of the flattened GEMM
#define FULL_KTILES 18          // 18 full 16-wide tiles + 1 partial (k=288..299)

__global__ __launch_bounds__(256, 2)
void rbf_wmma_kernel(const float* __restrict__ x,
                     const float* __restrict__ c,
                     const float* __restrict__ sigma,
                     float* __restrict__ out)
{
    const int lane  = threadIdx.x & 31;
    const int wave  = threadIdx.x >> 5;
    const int half  = lane >> 4;     // 0: lanes 0-15, 1: lanes 16-31
    const int l15   = lane & 15;

    const int rowTile = blockIdx.x * 8 + wave;   // each wave: 16 rows x all K
    const int rowBase = rowTile * 16;

    // ---- A matrix (16x4 f32, 2 VGPRs) ----
    // lanes 0-15: v0=K0, v1=K1 for row M=l15 ; lanes 16-31: v0=K2, v1=K3(pad=1.0)
    const float* xr = x + (size_t)(rowBase + l15) * F_DIM;
    const float x0 = xr[0], x1 = xr[1], x2 = xr[2];
    v2f A;
    A.x = half ? x2   : x0;
    A.y = half ? 1.0f : x1;

    // ---- C matrix: ||x_row||^2 broadcast along each row ----
    // D/C layout: VGPR j, lanes 0-15 -> row j ; lanes 16-31 -> row j+8.
    const float x2row = x0 * x0 + x1 * x1 + x2 * x2;   // ||x_{rowBase+l15}||^2
    v8f C;
#pragma unroll
    for (int j = 0; j < 8; ++j)
        C[j] = __shfl(x2row, j + half * 8, 32);

    const float NLOG2E = -1.44269504088896340736f;     // exp(-t) = exp2(-t*log2e)

    // Single per-lane store base; all stores become immediate-offset off this.
    float* outBase = out + (size_t)(rowBase + half * 8) * K_DIM + l15;

    // ---- 18 full tiles: EXEC stays all-1s, fully unrolled ----
#pragma unroll
    for (int kt = 0; kt < FULL_KTILES; ++kt) {
        const int k = kt * 16 + l15;               // this lane's center/column
        const float* cr = c + (size_t)k * F_DIM;
        const float c0 = cr[0], c1 = cr[1], c2 = cr[2];
        const float sg = sigma[k];
        const float cn = c0 * c0 + c1 * c1 + c2 * c2;   // ||c_k||^2

        // B matrix (4x16 f32, 2 VGPRs), col n=l15:
        // lanes 0-15: v0=K0, v1=K1 ; lanes 16-31: v0=K2, v1=K3
        v2f Bm;
        Bm.x = half ? (-2.0f * c2) : (-2.0f * c0);
        Bm.y = half ? cn           : (-2.0f * c1);

        // d2[m][n] = ||x_m||^2 - 2 x_m . c_n + ||c_n||^2
        v8f d2 = __builtin_amdgcn_wmma_f32_16x16x4_f32(
            false, A, false, Bm, (short)0, C, false, false);

        // fast reciprocal: one v_rcp_f32 instead of the IEEE divide chain
        const float scale = NLOG2E * __builtin_amdgcn_rcpf(sg * sg);

#pragma unroll
        for (int j = 0; j < 8; ++j) {
            const float v = __builtin_amdgcn_exp2f(scale * d2[j]);
            __builtin_nontemporal_store(v, outBase + j * K_DIM + kt * 16);
        }
    }

    // ---- tail tile kt=18: columns k=288..303, only k<300 valid ----
    {
        const int k  = FULL_KTILES * 16 + l15;     // 288 + l15
        const int kc = (k < K_DIM) ? k : (K_DIM - 1);
        const float* cr = c + (size_t)kc * F_DIM;
        const float c0 = cr[0], c1 = cr[1], c2 = cr[2];
        const float sg = sigma[kc];
        const float cn = c0 * c0 + c1 * c1 + c2 * c2;

        v2f Bm;
        Bm.x = half ? (-2.0f * c2) : (-2.0f * c0);
        Bm.y = half ? cn           : (-2.0f * c1);

        v8f d2 = __builtin_amdgcn_wmma_f32_16x16x4_f32(
            false, A, false, Bm, (short)0, C, false, false);  // full EXEC here

        const float scale = NLOG2E * __builtin_amdgcn_rcpf(sg * sg);

        if (l15 < (K_DIM - FULL_KTILES * 16)) {    // only stores predicated
#pragma unroll
            for (int j = 0; j < 8; ++j) {
                const float v = __builtin_amdgcn_exp2f(scale * d2[j]);
                __builtin_nontemporal_store(v, outBase + j * K_DIM + FULL_KTILES * 16);
            }
        }
    }
}

extern "C" void kernel_launch(void* const* d_in, const int* in_sizes, int n_in,
                              void* d_out, int out_size, void* d_ws, size_t ws_size,
                              hipStream_t stream) {
    (void)in_sizes; (void)n_in; (void)d_ws; (void)ws_size; (void)out_size;
    const float* x     = (const float*)d_in[0];   // (32,1024,3)
    const float* c     = (const float*)d_in[1];   // (300,3)
    const float* sigma = (const float*)d_in[2];   // (300,1)
    float*       out   = (float*)d_out;           // (32,1024,300)

    const int rowTiles = ROWS / 16;               // 2048
    const int blocks   = rowTiles / 8;            // 256 blocks x 8 wave32
    rbf_wmma_kernel<<<blocks, 256, 0, stream>>>(x, c, sigma, out);
}